// DecompGrid_34617436406212
// MI455X (gfx1250) — compile-verified
//
#include <hip/hip_runtime.h>
#include <hip/hip_bf16.h>
#include <cstdint>

#define CCH   16
#define R3    128
#define R2    256

// ---------------- gfx1250 async global->LDS helpers ----------------
typedef int v4i __attribute__((vector_size(16)));
typedef __attribute__((address_space(1))) v4i* gv4p;   // global int4*
typedef __attribute__((address_space(3))) v4i* lv4p;   // LDS int4*

__device__ __forceinline__ void async_load_b128(const void* g, void* l) {
#if __has_builtin(__builtin_amdgcn_global_load_async_to_lds_b128)
  v4i* gg = (v4i*)g;   // drop const, generic pointer
  v4i* ll = (v4i*)l;
  __builtin_amdgcn_global_load_async_to_lds_b128((gv4p)gg, (lv4p)ll, 0, 0);
#else
  asm volatile("global_load_async_to_lds_b128 %0, %1, off"
               :: "v"((uint32_t)(uintptr_t)l), "v"(g) : "memory");
#endif
}

__device__ __forceinline__ void wait_async0() {
#if __has_builtin(__builtin_amdgcn_s_wait_asynccnt)
  __builtin_amdgcn_s_wait_asynccnt(0);
#else
  asm volatile("s_wait_asynccnt 0" ::: "memory");
#endif
}

// ---------------- layout transpose: (16, S) -> (S, 16) ----------------
__global__ __launch_bounds__(256) void transpose_c16(const float* __restrict__ in,
                                                     float* __restrict__ out, int S) {
  int s = blockIdx.x * blockDim.x + threadIdx.x;
  if (s >= S) return;
  float v[CCH];
#pragma unroll
  for (int c = 0; c < CCH; ++c)
    v[c] = __builtin_nontemporal_load(in + (size_t)c * S + s);  // source is dead after this
  float4* o = (float4*)(out + (size_t)s * CCH);
#pragma unroll
  for (int j = 0; j < 4; ++j)
    o[j] = make_float4(v[4 * j + 0], v[4 * j + 1], v[4 * j + 2], v[4 * j + 3]);
}

// gather 16 channels at flat cell index, fma into acc with weight w
template <bool TR>
__device__ __forceinline__ void gather16(const float* __restrict__ base, int cell,
                                         int S, float w, float acc[CCH]) {
  if (TR) {
    // (spatial, C) layout: one 64B cacheline per corner, 4x b128 loads
    const float4* g = (const float4*)(base + ((int64_t)cell << 4));
    union { float4 v[4]; float f[CCH]; } U;
    U.v[0] = g[0]; U.v[1] = g[1]; U.v[2] = g[2]; U.v[3] = g[3];
#pragma unroll
    for (int c = 0; c < CCH; ++c) acc[c] = fmaf(w, U.f[c], acc[c]);
  } else {
    // fallback: reference (C, spatial) layout, channel-strided
#pragma unroll
    for (int c = 0; c < CCH; ++c)
      acc[c] = fmaf(w, base[(int64_t)c * S + cell], acc[c]);
  }
}

template <bool TR>
__device__ __forceinline__ void sample2d(const float* __restrict__ P, float u, float v,
                                         float r[CCH]) {
  float fu = (u + 1.0f) * 0.5f * (R2 - 1);
  float fv = (v + 1.0f) * 0.5f * (R2 - 1);
  float u0f = floorf(fu), v0f = floorf(fv);
  float wu = fu - u0f, wv = fv - v0f;
  int u0 = ::min(::max((int)u0f, 0), R2 - 1); int u1 = ::min(u0 + 1, R2 - 1);
  int v0 = ::min(::max((int)v0f, 0), R2 - 1); int v1 = ::min(v0 + 1, R2 - 1);
#pragma unroll
  for (int c = 0; c < CCH; ++c) r[c] = 0.0f;
  gather16<TR>(P, v0 * R2 + u0, R2 * R2, (1.0f - wv) * (1.0f - wu), r);
  gather16<TR>(P, v0 * R2 + u1, R2 * R2, (1.0f - wv) * wu,          r);
  gather16<TR>(P, v1 * R2 + u0, R2 * R2, wv * (1.0f - wu),          r);
  gather16<TR>(P, v1 * R2 + u1, R2 * R2, wv * wu,                   r);
}

// ---------------- main sampling kernel ----------------
template <bool TR>
__global__ __launch_bounds__(256) void sample_kernel(
    const float* __restrict__ x,   const float* __restrict__ g3,
    const float* __restrict__ p01, const float* __restrict__ p02,
    const float* __restrict__ p12, float* __restrict__ out, int B) {
  __shared__ __align__(16) float xs[256 * 3];

  const int t = threadIdx.x;
  const int64_t b0 = (int64_t)blockIdx.x * 256;

  // Stage this block's 3KB coordinate slab into LDS with async b128 transfers
  // (ASYNCcnt path). 192 lanes x 16B = 3072B.
  if (t < 192 && (b0 * 3 + t * 4) < (int64_t)B * 3)
    async_load_b128(x + b0 * 3 + (int64_t)t * 4, &xs[t * 4]);
  wait_async0();
  __syncthreads();

  const int64_t b = b0 + t;
  if (b >= B) return;

  const float px = xs[3 * t + 0];
  const float py = xs[3 * t + 1];
  const float pz = xs[3 * t + 2];

  // ---- 3D trilinear ----
  float fx = (px + 1.0f) * 0.5f * (R3 - 1);
  float fy = (py + 1.0f) * 0.5f * (R3 - 1);
  float fz = (pz + 1.0f) * 0.5f * (R3 - 1);
  float x0f = floorf(fx), y0f = floorf(fy), z0f = floorf(fz);
  float wx = fx - x0f, wy = fy - y0f, wz = fz - z0f;
  int x0 = ::min(::max((int)x0f, 0), R3 - 1); int x1 = ::min(x0 + 1, R3 - 1);
  int y0 = ::min(::max((int)y0f, 0), R3 - 1); int y1 = ::min(y0 + 1, R3 - 1);
  int z0 = ::min(::max((int)z0f, 0), R3 - 1); int z1 = ::min(z0 + 1, R3 - 1);

  float acc[CCH];
#pragma unroll
  for (int c = 0; c < CCH; ++c) acc[c] = 0.0f;

  const int S3 = R3 * R3 * R3;
  gather16<TR>(g3, (z0 * R3 + y0) * R3 + x0, S3, (1.0f - wz) * (1.0f - wy) * (1.0f - wx), acc);
  gather16<TR>(g3, (z0 * R3 + y0) * R3 + x1, S3, (1.0f - wz) * (1.0f - wy) * wx,          acc);
  gather16<TR>(g3, (z0 * R3 + y1) * R3 + x0, S3, (1.0f - wz) * wy * (1.0f - wx),          acc);
  gather16<TR>(g3, (z0 * R3 + y1) * R3 + x1, S3, (1.0f - wz) * wy * wx,                   acc);
  gather16<TR>(g3, (z1 * R3 + y0) * R3 + x0, S3, wz * (1.0f - wy) * (1.0f - wx),          acc);
  gather16<TR>(g3, (z1 * R3 + y0) * R3 + x1, S3, wz * (1.0f - wy) * wx,                   acc);
  gather16<TR>(g3, (z1 * R3 + y1) * R3 + x0, S3, wz * wy * (1.0f - wx),                   acc);
  gather16<TR>(g3, (z1 * R3 + y1) * R3 + x1, S3, wz * wy * wx,                            acc);

  // ---- three bilinear plane factors ----
  float pl[CCH];
  sample2d<TR>(p01, px, py, pl);
#pragma unroll
  for (int c = 0; c < CCH; ++c) acc[c] *= pl[c];
  sample2d<TR>(p02, px, pz, pl);
#pragma unroll
  for (int c = 0; c < CCH; ++c) acc[c] *= pl[c];
  sample2d<TR>(p12, py, pz, pl);
#pragma unroll
  for (int c = 0; c < CCH; ++c) acc[c] *= pl[c];

  // Output (C, B): coalesced across lanes per channel. Non-temporal so the
  // 64MB stream does not evict the 140MB of L2-resident tables.
#pragma unroll
  for (int c = 0; c < CCH; ++c)
    __builtin_nontemporal_store(acc[c], out + (size_t)c * B + b);
}

extern "C" void kernel_launch(void* const* d_in, const int* in_sizes, int n_in,
                              void* d_out, int out_size, void* d_ws, size_t ws_size,
                              hipStream_t stream) {
  const float* x   = (const float*)d_in[0];
  const float* g3  = (const float*)d_in[1];
  const float* p01 = (const float*)d_in[2];
  const float* p02 = (const float*)d_in[3];
  const float* p12 = (const float*)d_in[4];
  float* out = (float*)d_out;

  const int B = in_sizes[0] / 3;
  const size_t S3 = (size_t)R3 * R3 * R3;        // 2,097,152
  const size_t S2 = (size_t)R2 * R2;             // 65,536
  const size_t need = (S3 + 3 * S2) * CCH * sizeof(float);  // ~140 MB

  const int blocks = (B + 255) / 256;

  if (ws_size >= need) {
    float* g3t  = (float*)d_ws;
    float* p01t = g3t  + S3 * CCH;
    float* p02t = p01t + S2 * CCH;
    float* p12t = p02t + S2 * CCH;
    transpose_c16<<<(int)((S3 + 255) / 256), 256, 0, stream>>>(g3,  g3t,  (int)S3);
    transpose_c16<<<(int)((S2 + 255) / 256), 256, 0, stream>>>(p01, p01t, (int)S2);
    transpose_c16<<<(int)((S2 + 255) / 256), 256, 0, stream>>>(p02, p02t, (int)S2);
    transpose_c16<<<(int)((S2 + 255) / 256), 256, 0, stream>>>(p12, p12t, (int)S2);
    sample_kernel<true><<<blocks, 256, 0, stream>>>(x, g3t, p01t, p02t, p12t, out, B);
  } else {
    // workspace too small: correct fallback in the original channel-major layout
    sample_kernel<false><<<blocks, 256, 0, stream>>>(x, g3, p01, p02, p12, out, B);
  }
}